// AttentionCTSF_19997367730269
// MI455X (gfx1250) — compile-verified
//
#include <hip/hip_runtime.h>
#include <hip/hip_bf16.h>

// ---------------------------------------------------------------------------
// AttentionCTSF on MI455X (gfx1250): fp32 end-to-end via V_WMMA_F32_16X16X4_F32.
// Bandwidth-bound pipeline (~1.5 GB traffic vs ~28 GFLOP), so fp32 WMMA is the
// right precision/perf point; every GEMM stage emits v_wmma.
// ---------------------------------------------------------------------------

typedef float v2f __attribute__((ext_vector_type(2)));
typedef float v8f __attribute__((ext_vector_type(8)));

#define BB_   2
#define CC_   64
#define TT_   10
#define HH_   128
#define WW_   128
#define HW_   (HH_ * WW_)      // 16384
#define THW_  (TT_ * HW_)      // 163840
#define C3_   (3 * CC_)        // 192
#define C2_   24
#define L_    80               // SHUFFLE_G * t
#define HEADS_ 8
#define CT_   80

__device__ __forceinline__ v8f wmma4(v2f a, v2f b, v8f c) {
  // D = A(16x4,f32) * B(4x16,f32) + C(16x16,f32)
  return __builtin_amdgcn_wmma_f32_16x16x4_f32(
      /*neg_a=*/false, a, /*neg_b=*/false, b,
      /*c_mod=*/(short)0, c, /*reuse_a=*/false, /*reuse_b=*/false);
}

// ---------------- K1: pointwise conv  y1[b,o,t,p] = sum_c W1[o,c] x[b,c,t,p]
__global__ void k1_pointwise(const float* __restrict__ x,
                             const float* __restrict__ W1,
                             float* __restrict__ y1) {
  const int lane = threadIdx.x & 31;
  const int wv   = threadIdx.x >> 5;
  const int mn   = lane & 15;          // M index (A) / N index (B,D)
  const int kh   = (lane >> 4) * 2;    // K half offset
  const int n0   = blockIdx.x * 128 + wv * 16;   // pixel tile
  const int m0   = blockIdx.y * 16;              // out-channel tile
  const int bt   = blockIdx.z;
  const int bb   = bt / TT_, tt = bt % TT_;
  const float* xb = x + (size_t)bb * CC_ * THW_ + (size_t)tt * HW_;
  v8f acc = {};
  for (int k0 = 0; k0 < CC_; k0 += 4) {
    v2f a, bf;
    a.x  = W1[(m0 + mn) * CC_ + k0 + kh];
    a.y  = W1[(m0 + mn) * CC_ + k0 + kh + 1];
    bf.x = xb[(size_t)(k0 + kh)     * THW_ + n0 + mn];
    bf.y = xb[(size_t)(k0 + kh + 1) * THW_ + n0 + mn];
    acc = wmma4(a, bf, acc);
  }
  float* yb = y1 + (size_t)bb * C3_ * THW_ + (size_t)tt * HW_;
#pragma unroll
  for (int r = 0; r < 8; ++r) {
    int m = r + (lane >> 4) * 8;
    yb[(size_t)(m0 + m) * THW_ + n0 + mn] = acc[r];
  }
}

// ---------------- K2: depthwise (1,3,3) conv, zero pad (0,1,1)
__global__ void k2_depthwise(const float* __restrict__ y1,
                             const float* __restrict__ Wdw,
                             float* __restrict__ ydw) {
  size_t idx = (size_t)blockIdx.x * blockDim.x + threadIdx.x;
  const size_t total = (size_t)BB_ * C3_ * TT_ * HW_;
  if (idx >= total) return;
  int p  = (int)(idx % HW_);
  size_t r = idx / HW_;
  int tt = (int)(r % TT_); r /= TT_;
  int ch = (int)(r % C3_);
  (void)tt;
  int hh = p / WW_, ww = p % WW_;
  const float* src = y1 + (idx - (size_t)p);   // base of (b,ch,t) plane
  const float* wk  = Wdw + ch * 9;
  float s = 0.f;
#pragma unroll
  for (int dy = -1; dy <= 1; ++dy) {
    int yy = hh + dy;
    if ((unsigned)yy >= (unsigned)HH_) continue;
#pragma unroll
    for (int dx = -1; dx <= 1; ++dx) {
      int xx = ww + dx;
      if ((unsigned)xx >= (unsigned)WW_) continue;
      s += wk[(dy + 1) * 3 + (dx + 1)] * src[yy * WW_ + xx];
    }
  }
  ydw[idx] = s;
}

// ---------------- K3: shuffle + W_lin (80x80) + bias + unshuffle -> qkv
// qkv[b, c2, o, p] = sum_i Wlin[o,i] * ydw[b, (i/10)*24+c2, i%10, p] + blin[o]
__global__ void k3_linear(const float* __restrict__ ydw,
                          const float* __restrict__ Wlin,
                          const float* __restrict__ blin,
                          float* __restrict__ qkv) {
  const int lane = threadIdx.x & 31;
  const int wv   = threadIdx.x >> 5;
  const int mn   = lane & 15;
  const int kh   = (lane >> 4) * 2;
  const int n0   = blockIdx.x * 128 + wv * 16;
  const int m0   = blockIdx.y * 16;
  const int bc   = blockIdx.z;
  const int bb   = bc / C2_, c2 = bc % C2_;
  v8f acc = {};
  for (int k0 = 0; k0 < L_; k0 += 4) {
    int i0 = k0 + kh, i1 = i0 + 1;
    v2f a, bf;
    a.x = Wlin[(m0 + mn) * L_ + i0];
    a.y = Wlin[(m0 + mn) * L_ + i1];
    int ch0 = (i0 / TT_) * C2_ + c2, t0 = i0 % TT_;
    int ch1 = (i1 / TT_) * C2_ + c2, t1 = i1 % TT_;
    bf.x = ydw[((size_t)(bb * C3_ + ch0) * TT_ + t0) * HW_ + n0 + mn];
    bf.y = ydw[((size_t)(bb * C3_ + ch1) * TT_ + t1) * HW_ + n0 + mn];
    acc = wmma4(a, bf, acc);
  }
#pragma unroll
  for (int r = 0; r < 8; ++r) {
    int m = r + (lane >> 4) * 8;
    qkv[((size_t)(bb * C2_ + c2) * L_ + m0 + m) * HW_ + n0 + mn] =
        acc[r] + blin[m0 + m];
  }
}

// ---------------- K4: row L2 norms for q (c2 in [0,8)) and k (c2 in [8,16))
__global__ void k4_norms(const float* __restrict__ qkv, float* __restrict__ nrm) {
  int id  = blockIdx.x;
  int row = id % CT_; id /= CT_;
  int c2  = id % 16;  id /= 16;
  int bb  = id;
  const float* src = qkv + ((size_t)(bb * C2_ + c2) * L_ + row) * HW_;
  float s = 0.f;
  for (int p = threadIdx.x; p < HW_; p += blockDim.x) {
    float v = src[p];
    s += v * v;
  }
  __shared__ float red[256];
  red[threadIdx.x] = s;
  __syncthreads();
  for (int st = 128; st > 0; st >>= 1) {
    if (threadIdx.x < st) red[threadIdx.x] += red[threadIdx.x + st];
    __syncthreads();
  }
  if (threadIdx.x == 0)
    nrm[(bb * 16 + c2) * CT_ + row] = fmaxf(sqrtf(red[0]), 1e-12f);
}

// ---------------- K5: S = (Q K^T) / (nq nk) * temp   (K = 16384, 8-way split)
__global__ void k5_scores(const float* __restrict__ qkv,
                          const float* __restrict__ nrm,
                          const float* __restrict__ temp,
                          float* __restrict__ attnS) {
  const int lane = threadIdx.x & 31;
  const int wv   = threadIdx.x >> 5;   // K-split 0..7
  const int mn   = lane & 15;
  const int kh   = (lane >> 4) * 2;
  const int j0   = blockIdx.x * 16;
  const int i0   = blockIdx.y * 16;
  const int bh   = blockIdx.z;
  const int bb   = bh / HEADS_, hd = bh % HEADS_;
  const float* q = qkv + (size_t)(bb * C2_ + hd)     * CT_ * HW_;
  const float* k = qkv + (size_t)(bb * C2_ + hd + 8) * CT_ * HW_;
  v8f acc = {};
  const int kbeg = wv * (HW_ / 8), kend = kbeg + (HW_ / 8);
  for (int kk = kbeg; kk < kend; kk += 4) {
    v2f a, bf;
    a.x  = q[(size_t)(i0 + mn) * HW_ + kk + kh];
    a.y  = q[(size_t)(i0 + mn) * HW_ + kk + kh + 1];
    bf.x = k[(size_t)(j0 + mn) * HW_ + kk + kh];       // B[p, j] = k[j, p]
    bf.y = k[(size_t)(j0 + mn) * HW_ + kk + kh + 1];
    acc = wmma4(a, bf, acc);
  }
  __shared__ float red[8][16][16];
#pragma unroll
  for (int r = 0; r < 8; ++r) {
    int m = r + (lane >> 4) * 8;
    red[wv][m][mn] = acc[r];
  }
  __syncthreads();
  int m = threadIdx.x >> 4, n = threadIdx.x & 15;
  float s = 0.f;
#pragma unroll
  for (int w8 = 0; w8 < 8; ++w8) s += red[w8][m][n];
  float nq = nrm[(bb * 16 + hd)     * CT_ + i0 + m];
  float nk = nrm[(bb * 16 + hd + 8) * CT_ + j0 + n];
  s = s / (nq * nk) * temp[hd];
  attnS[((size_t)(bb * HEADS_ + hd) * CT_ + i0 + m) * CT_ + j0 + n] = s;
}

// ---------------- K6: softmax along rows of 80
__global__ void k6_softmax(float* __restrict__ attnS) {
  int bh  = blockIdx.x;
  int row = threadIdx.x;
  if (row >= CT_) return;
  float* r = attnS + ((size_t)bh * CT_ + row) * CT_;
  float mx = -3.4e38f;
  for (int j = 0; j < CT_; ++j) mx = fmaxf(mx, r[j]);
  float s = 0.f;
  for (int j = 0; j < CT_; ++j) { float e = expf(r[j] - mx); r[j] = e; s += e; }
  float inv = 1.f / s;
  for (int j = 0; j < CT_; ++j) r[j] *= inv;
}

// ---------------- K7: out = attn @ V  (M=80, K=80, N=16384)
__global__ void k7_av(const float* __restrict__ attnS,
                      const float* __restrict__ qkv,
                      float* __restrict__ aout) {
  const int lane = threadIdx.x & 31;
  const int wv   = threadIdx.x >> 5;
  const int mn   = lane & 15;
  const int kh   = (lane >> 4) * 2;
  const int n0   = blockIdx.x * 128 + wv * 16;
  const int m0   = blockIdx.y * 16;
  const int bh   = blockIdx.z;
  const int bb   = bh / HEADS_, hd = bh % HEADS_;
  const float* A = attnS + (size_t)(bb * HEADS_ + hd) * CT_ * CT_;
  const float* V = qkv   + (size_t)(bb * C2_ + hd + 16) * CT_ * HW_;
  v8f acc = {};
  for (int k0 = 0; k0 < CT_; k0 += 4) {
    v2f a, bf;
    a.x  = A[(m0 + mn) * CT_ + k0 + kh];
    a.y  = A[(m0 + mn) * CT_ + k0 + kh + 1];
    bf.x = V[(size_t)(k0 + kh)     * HW_ + n0 + mn];
    bf.y = V[(size_t)(k0 + kh + 1) * HW_ + n0 + mn];
    acc = wmma4(a, bf, acc);
  }
  // [b, head, ct, p] flat == [b, c, t, p] flat (head*80+ct == c*10+t)
  float* dst = aout + (size_t)(bb * HEADS_ + hd) * CT_ * HW_;
#pragma unroll
  for (int r = 0; r < 8; ++r) {
    int m = r + (lane >> 4) * 8;
    dst[(size_t)(m0 + m) * HW_ + n0 + mn] = acc[r];
  }
}

// ---------------- K8: final pointwise conv  z = W_out (64x64) * aout
__global__ void k8_out(const float* __restrict__ aout,
                       const float* __restrict__ Wout,
                       float* __restrict__ out) {
  const int lane = threadIdx.x & 31;
  const int wv   = threadIdx.x >> 5;
  const int mn   = lane & 15;
  const int kh   = (lane >> 4) * 2;
  const int n0   = blockIdx.x * 128 + wv * 16;
  const int m0   = blockIdx.y * 16;
  const int bt   = blockIdx.z;
  const int bb   = bt / TT_, tt = bt % TT_;
  const float* src = aout + (size_t)bb * CC_ * THW_ + (size_t)tt * HW_;
  v8f acc = {};
  for (int k0 = 0; k0 < CC_; k0 += 4) {
    v2f a, bf;
    a.x  = Wout[(m0 + mn) * CC_ + k0 + kh];
    a.y  = Wout[(m0 + mn) * CC_ + k0 + kh + 1];
    bf.x = src[(size_t)(k0 + kh)     * THW_ + n0 + mn];
    bf.y = src[(size_t)(k0 + kh + 1) * THW_ + n0 + mn];
    acc = wmma4(a, bf, acc);
  }
  float* dst = out + (size_t)bb * CC_ * THW_ + (size_t)tt * HW_;
#pragma unroll
  for (int r = 0; r < 8; ++r) {
    int m = r + (lane >> 4) * 8;
    dst[(size_t)(m0 + m) * THW_ + n0 + mn] = acc[r];
  }
}

extern "C" void kernel_launch(void* const* d_in, const int* in_sizes, int n_in,
                              void* d_out, int out_size, void* d_ws, size_t ws_size,
                              hipStream_t stream) {
  const float* x    = (const float*)d_in[0];
  const float* W1   = (const float*)d_in[1];
  const float* Wdw  = (const float*)d_in[2];
  const float* Wlin = (const float*)d_in[3];
  const float* blin = (const float*)d_in[4];
  const float* temp = (const float*)d_in[5];
  const float* Wout = (const float*)d_in[6];
  float* out = (float*)d_out;
  float* ws  = (float*)d_ws;

  const size_t BIG = (size_t)BB_ * C3_ * THW_;   // 62,914,560 floats (252 MB)
  float* y1    = ws;               // K1 out; region reused as qkv by K3
  float* ydw   = ws + BIG;         // K2 out; region reused as aout by K7
  float* qkv   = y1;
  float* aout  = ydw;
  float* nrm   = ws + 2 * BIG;                       // B*16*CT floats
  float* attnS = nrm + (size_t)BB_ * 16 * CT_;       // B*HEADS*CT*CT floats
  (void)in_sizes; (void)n_in; (void)out_size; (void)ws_size;

  // K1: pointwise 192x64 GEMM
  k1_pointwise<<<dim3(HW_ / 128, C3_ / 16, BB_ * TT_), 256, 0, stream>>>(x, W1, y1);
  // K2: depthwise 3x3
  {
    size_t total = (size_t)BB_ * C3_ * TT_ * HW_;
    k2_depthwise<<<(unsigned)((total + 255) / 256), 256, 0, stream>>>(y1, Wdw, ydw);
  }
  // K3: shuffle + 80x80 linear + bias (48 GEMMs)
  k3_linear<<<dim3(HW_ / 128, L_ / 16, BB_ * C2_), 256, 0, stream>>>(ydw, Wlin, blin, qkv);
  // K4: q/k row norms
  k4_norms<<<BB_ * 16 * CT_, 256, 0, stream>>>(qkv, nrm);
  // K5: scores with fused l2norm scaling + temperature
  k5_scores<<<dim3(CT_ / 16, CT_ / 16, BB_ * HEADS_), 256, 0, stream>>>(qkv, nrm, temp, attnS);
  // K6: softmax
  k6_softmax<<<BB_ * HEADS_, 128, 0, stream>>>(attnS);
  // K7: attn @ V
  k7_av<<<dim3(HW_ / 128, CT_ / 16, BB_ * HEADS_), 256, 0, stream>>>(attnS, qkv, aout);
  // K8: output 64x64 pointwise conv
  k8_out<<<dim3(HW_ / 128, CC_ / 16, BB_ * TT_), 256, 0, stream>>>(aout, Wout, out);
}